// PaiNNInteractionBlock_31559419691312
// MI455X (gfx1250) — compile-verified
//
#include <hip/hip_runtime.h>

typedef __attribute__((ext_vector_type(2))) float v2f;
typedef __attribute__((ext_vector_type(8))) float v8f;

#define NN 20000
#define NE 600000
#define D  128
#define ESZ 20
#define CUT 5.0f

// D = A(16x4) * B(4x16) + C, fp32 WMMA.
// A fragment: lane(0-15) rows M=0..15 K={k0,k0+1}; lane(16-31) rows M=0..15 K={k0+2,k0+3}
// B fragment: lane(0-15) cols N=0..15 K={k0,k0+1}; lane(16-31) cols N=0..15 K={k0+2,k0+3}
// C/D: vgpr r, lane n: row r + 8*(lane>=16), col n&15
__device__ __forceinline__ v8f wmma_f32(v2f a, v2f b, v8f c) {
  return __builtin_amdgcn_wmma_f32_16x16x4_f32(false, a, false, b, (short)0, c, false, false);
}

__device__ __forceinline__ float silu_f(float x) { return x / (1.0f + __expf(-x)); }

// ---------------------------------------------------------------------------
// K1: scalar_output = silu(ns @ Wm1 + bm1) @ Wm2 + bm2     (16 nodes / block)
// ---------------------------------------------------------------------------
__global__ void __launch_bounds__(128) node_mlp_kernel(
    const float* __restrict__ ns, const float* __restrict__ Wm1,
    const float* __restrict__ bm1, const float* __restrict__ Wm2,
    const float* __restrict__ bm2, float* __restrict__ so) {
  __shared__ float hbuf[16][D + 4];
  const int tid = threadIdx.x;
  const int wave = tid >> 5, lane = tid & 31;
  const int half = lane >> 4, lrow = lane & 15;
  const int n0 = blockIdx.x * 16;

  // Stage 1: h (16x128), wave owns cols [wave*32, wave*32+32)
  const float* arow = ns + (size_t)(n0 + lrow) * D;
  for (int sub = 0; sub < 2; ++sub) {
    const int c0 = wave * 32 + sub * 16;
    v8f acc = {};
    for (int k0 = 0; k0 < D; k0 += 4) {
      const int k = k0 + 2 * half;
      v2f a; a[0] = arow[k]; a[1] = arow[k + 1];
      const float* bp = Wm1 + (size_t)k * D + c0 + lrow;
      v2f b; b[0] = bp[0]; b[1] = bp[D];
      acc = wmma_f32(a, b, acc);
    }
    const float bias = bm1[c0 + lrow];
    for (int r = 0; r < 8; ++r)
      hbuf[r + 8 * half][c0 + lrow] = silu_f(acc[r] + bias);
  }
  __syncthreads();

  // Stage 2: so (16x384), wave owns 6 consecutive column tiles
  for (int j = 0; j < 6; ++j) {
    const int c0 = (wave * 6 + j) * 16;
    v8f acc = {};
    for (int k0 = 0; k0 < D; k0 += 4) {
      const int k = k0 + 2 * half;
      v2f a; a[0] = hbuf[lrow][k]; a[1] = hbuf[lrow][k + 1];
      const float* bp = Wm2 + (size_t)k * 384 + c0 + lrow;
      v2f b; b[0] = bp[0]; b[1] = bp[384];
      acc = wmma_f32(a, b, acc);
    }
    const float bias = bm2[c0 + lrow];
    for (int r = 0; r < 8; ++r)
      so[(size_t)(n0 + r + 8 * half) * 384 + c0 + lrow] = acc[r] + bias;
  }
}

// ---------------------------------------------------------------------------
// K2: per 16 edges: filter GEMM (K=20), *fc, *scalar_output[src], form
// messages, atomic scatter-add into s_acc / v_acc (pre-seeded with node state)
// ---------------------------------------------------------------------------
__global__ void __launch_bounds__(128) edge_kernel(
    const float* __restrict__ es, const float* __restrict__ ev,
    const float* __restrict__ en, const int* __restrict__ eidx,
    const float* __restrict__ Wf, const float* __restrict__ bf,
    const float* __restrict__ so, const float* __restrict__ nv,
    float* __restrict__ s_acc, float* __restrict__ v_acc) {
  __shared__ int sS[16], sD[16];
  __shared__ float sFC[16], sEV[16][3];
  const int tid = threadIdx.x;
  const int wave = tid >> 5, lane = tid & 31;
  const int half = lane >> 4, lrow = lane & 15;
  const int e0 = blockIdx.x * 16;

  if (tid < 16) {
    const int e = e0 + tid;
    sS[tid] = eidx[2 * e];
    sD[tid] = eidx[2 * e + 1];
    const float nrm = en[e];
    sFC[tid] = (nrm < CUT) ? 0.5f * (__cosf(3.14159265358979f * nrm / CUT) + 1.0f)
                           : 0.0f;
    sEV[tid][0] = ev[3 * e];
    sEV[tid][1] = ev[3 * e + 1];
    sEV[tid][2] = ev[3 * e + 2];
  }
  __syncthreads();

  const float* arow = es + (size_t)(e0 + lrow) * ESZ;
  // column-tile assignment t = wave + 4*jj keeps gate_nodes(c) / gate_edges(c)
  // / messages_scalar(c) slices on the same wave.
  for (int jj = 0; jj < 2; ++jj) {
    const int c = (wave + 4 * jj) * 16 + lrow;
    v8f agn = {}, age = {}, ams = {};
    for (int k0 = 0; k0 < ESZ; k0 += 4) {
      const int k = k0 + 2 * half;
      v2f a; a[0] = arow[k]; a[1] = arow[k + 1];
      const float* b0 = Wf + (size_t)k * 384 + c;
      v2f b1; b1[0] = b0[0];   b1[1] = b0[384];
      v2f b2; b2[0] = b0[128]; b2[1] = b0[512];
      v2f b3; b3[0] = b0[256]; b3[1] = b0[640];
      agn = wmma_f32(a, b1, agn);   // shared A fragment, 3 WMMAs
      age = wmma_f32(a, b2, age);
      ams = wmma_f32(a, b3, ams);
    }
    const float zgn = bf[c], zge = bf[c + 128], zms = bf[c + 256];
    for (int r = 0; r < 8; ++r) {
      const int row = r + 8 * half;
      const float fc = sFC[row];
      const int s = sS[row], d = sD[row];
      const float* sor = so + (size_t)s * 384;
      const float fgn = (agn[r] + zgn) * fc * sor[c];
      const float fge = (age[r] + zge) * fc * sor[c + 128];
      const float fms = (ams[r] + zms) * fc * sor[c + 256];
      atomicAdd(&s_acc[(size_t)d * D + c], fms);
      const float* nvr = nv + (size_t)s * 3 * D + c;
      float* var = v_acc + (size_t)d * 3 * D + c;
      for (int ax = 0; ax < 3; ++ax)
        atomicAdd(&var[ax * D], fgn * nvr[ax * D] + fge * sEV[row][ax]);
    }
  }
}

// ---------------------------------------------------------------------------
// K3: Uv = v@WU (stored), vsq = sum_ax Vv^2, ip = sum_ax Uv*Vv
// ---------------------------------------------------------------------------
__global__ void __launch_bounds__(128) equivariant_kernel(
    const float* __restrict__ v_pre, const float* __restrict__ WU,
    const float* __restrict__ WV, float* __restrict__ Uv,
    float* __restrict__ vsq, float* __restrict__ ip) {
  const int tid = threadIdx.x;
  const int wave = tid >> 5, lane = tid & 31;
  const int half = lane >> 4, lrow = lane & 15;
  const int n0 = blockIdx.x * 16;

  for (int sub = 0; sub < 2; ++sub) {
    const int c = wave * 32 + sub * 16 + lrow;
    v8f vsqa = {}, ipa = {};
    for (int ax = 0; ax < 3; ++ax) {
      v8f ua = {}, va = {};
      const float* arow = v_pre + ((size_t)(n0 + lrow) * 3 + ax) * D;
      for (int k0 = 0; k0 < D; k0 += 4) {
        const int k = k0 + 2 * half;
        v2f a; a[0] = arow[k]; a[1] = arow[k + 1];
        const float* bu = WU + (size_t)k * D + c;
        const float* bv = WV + (size_t)k * D + c;
        v2f fu; fu[0] = bu[0]; fu[1] = bu[D];
        v2f fv; fv[0] = bv[0]; fv[1] = bv[D];
        ua = wmma_f32(a, fu, ua);     // shared A fragment, 2 WMMAs
        va = wmma_f32(a, fv, va);
      }
      vsqa += va * va;
      ipa  += ua * va;
      for (int r = 0; r < 8; ++r)
        Uv[((size_t)(n0 + r + 8 * half) * 3 + ax) * D + c] = ua[r];
    }
    for (int r = 0; r < 8; ++r) {
      const size_t o = (size_t)(n0 + r + 8 * half) * D + c;
      vsq[o] = vsqa[r];
      ip[o]  = ipa[r];
    }
  }
}

// ---------------------------------------------------------------------------
// K4: a = silu([s|vsq]@Wa1+ba1)@Wa2+ba2; fused final update into d_out
// ---------------------------------------------------------------------------
__global__ void __launch_bounds__(128) update_kernel(
    const float* __restrict__ s_pre, const float* __restrict__ v_pre,
    const float* __restrict__ vsq, const float* __restrict__ ip,
    const float* __restrict__ Uv, const float* __restrict__ Wa1,
    const float* __restrict__ ba1, const float* __restrict__ Wa2,
    const float* __restrict__ ba2, float* __restrict__ s_out,
    float* __restrict__ v_out) {
  __shared__ float hbuf[16][D + 4];
  const int tid = threadIdx.x;
  const int wave = tid >> 5, lane = tid & 31;
  const int half = lane >> 4, lrow = lane & 15;
  const int n0 = blockIdx.x * 16;

  // Stage A: h = silu(concat(s_pre, vsq) @ Wa1 + ba1), K = 256
  const float* arow0 = s_pre + (size_t)(n0 + lrow) * D;
  const float* arow1 = vsq + (size_t)(n0 + lrow) * D;
  for (int sub = 0; sub < 2; ++sub) {
    const int c0 = wave * 32 + sub * 16;
    v8f acc = {};
    for (int k0 = 0; k0 < 256; k0 += 4) {
      const int k = k0 + 2 * half;
      v2f a;
      if (k0 < D) { a[0] = arow0[k];     a[1] = arow0[k + 1]; }
      else        { a[0] = arow1[k - D]; a[1] = arow1[k - D + 1]; }
      const float* bp = Wa1 + (size_t)k * D + c0 + lrow;
      v2f b; b[0] = bp[0]; b[1] = bp[D];
      acc = wmma_f32(a, b, acc);
    }
    const float bias = ba1[c0 + lrow];
    for (int r = 0; r < 8; ++r)
      hbuf[r + 8 * half][c0 + lrow] = silu_f(acc[r] + bias);
  }
  __syncthreads();

  // Stage B+C: a = h @ Wa2 + ba2, fused final elementwise update
  for (int jj = 0; jj < 2; ++jj) {
    const int c = (wave + 4 * jj) * 16 + lrow;
    v8f ass = {}, asv = {}, avv = {};
    for (int k0 = 0; k0 < D; k0 += 4) {
      const int k = k0 + 2 * half;
      v2f a; a[0] = hbuf[lrow][k]; a[1] = hbuf[lrow][k + 1];
      const float* b0 = Wa2 + (size_t)k * 384 + c;
      v2f b1; b1[0] = b0[0];   b1[1] = b0[384];
      v2f b2; b2[0] = b0[128]; b2[1] = b0[512];
      v2f b3; b3[0] = b0[256]; b3[1] = b0[640];
      ass = wmma_f32(a, b1, ass);
      asv = wmma_f32(a, b2, asv);
      avv = wmma_f32(a, b3, avv);
    }
    const float z1 = ba2[c], z2 = ba2[c + 128], z3 = ba2[c + 256];
    for (int r = 0; r < 8; ++r) {
      const int n = n0 + r + 8 * half;
      const size_t os = (size_t)n * D + c;
      const float a_ss = ass[r] + z1;
      const float a_sv = asv[r] + z2;
      const float a_vv = avv[r] + z3;
      s_out[os] = s_pre[os] + a_ss + a_sv * ip[os];
      for (int ax = 0; ax < 3; ++ax) {
        const size_t ov = ((size_t)n * 3 + ax) * D + c;
        v_out[ov] = v_pre[ov] + a_vv * Uv[ov];
      }
    }
  }
}

// ---------------------------------------------------------------------------
extern "C" void kernel_launch(void* const* d_in, const int* in_sizes, int n_in,
                              void* d_out, int out_size, void* d_ws,
                              size_t ws_size, hipStream_t stream) {
  const float* ns   = (const float*)d_in[0];
  const float* nv   = (const float*)d_in[1];
  const float* es   = (const float*)d_in[2];
  const float* ev   = (const float*)d_in[3];
  const float* en   = (const float*)d_in[4];
  const int*   eidx = (const int*)d_in[5];
  const float* Wf   = (const float*)d_in[6];
  const float* bf   = (const float*)d_in[7];
  const float* Wm1  = (const float*)d_in[8];
  const float* bm1  = (const float*)d_in[9];
  const float* Wm2  = (const float*)d_in[10];
  const float* bm2  = (const float*)d_in[11];
  const float* WU   = (const float*)d_in[12];
  const float* WV   = (const float*)d_in[13];
  const float* Wa1  = (const float*)d_in[14];
  const float* ba1  = (const float*)d_in[15];
  const float* Wa2  = (const float*)d_in[16];
  const float* ba2  = (const float*)d_in[17];

  float* ws = (float*)d_ws;
  float* so    = ws;                           // N*384
  float* s_pre = so    + (size_t)NN * 384;     // N*128
  float* v_pre = s_pre + (size_t)NN * 128;     // N*384
  float* Uv    = v_pre + (size_t)NN * 384;     // N*384
  float* vsq   = Uv    + (size_t)NN * 384;     // N*128
  float* ipb   = vsq   + (size_t)NN * 128;     // N*128

  float* s_out = (float*)d_out;
  float* v_out = s_out + (size_t)NN * 128;

  // Seed accumulators with node states (graph-capture-safe async D2D).
  hipMemcpyAsync(s_pre, ns, (size_t)NN * 128 * sizeof(float),
                 hipMemcpyDeviceToDevice, stream);
  hipMemcpyAsync(v_pre, nv, (size_t)NN * 384 * sizeof(float),
                 hipMemcpyDeviceToDevice, stream);

  node_mlp_kernel<<<NN / 16, 128, 0, stream>>>(ns, Wm1, bm1, Wm2, bm2, so);
  edge_kernel<<<NE / 16, 128, 0, stream>>>(es, ev, en, eidx, Wf, bf, so, nv,
                                           s_pre, v_pre);
  equivariant_kernel<<<NN / 16, 128, 0, stream>>>(v_pre, WU, WV, Uv, vsq, ipb);
  update_kernel<<<NN / 16, 128, 0, stream>>>(s_pre, v_pre, vsq, ipb, Uv, Wa1,
                                             ba1, Wa2, ba2, s_out, v_out);
}